// HMM_87462714016457
// MI455X (gfx1250) — compile-verified
//
#include <hip/hip_runtime.h>

#define N_BINS 10
#define WIDTH  128
#define LENGTH 256
#define BATCH  4096
#define STEPS  (LENGTH - 1)      // 255
#define ROWS   64                // batch rows per workgroup
#define BLOCK  256               // 8 waves (wave32)
#define PAD    4
#define WROW   (WIDTH + PAD)     // 132 floats per padded LDS row
#define CPY_IT ((WIDTH * WIDTH) / (BLOCK * 4))   // 16 b128 async ops per thread/tile

typedef __attribute__((ext_vector_type(2))) float v2f;
typedef __attribute__((ext_vector_type(8))) float v8f;

// Raw-transcendental exp/log: inputs are provably outside the denorm/inf guard
// regions, so skip ocml's fixup sequences (v_exp_f32 / v_log_f32 only).
__device__ __forceinline__ float fast_exp(float x) {
#if __has_builtin(__builtin_amdgcn_exp2f)
  return __builtin_amdgcn_exp2f(x * 1.44269504088896340736f);
#else
  return __expf(x);
#endif
}
__device__ __forceinline__ float fast_log(float x) {
#if __has_builtin(__builtin_amdgcn_logf)
  return __builtin_amdgcn_logf(x) * 0.69314718055994530942f;
#else
  return __logf(x);
#endif
}

// Async DMA of one 64KB weight tile into an LDS buffer (gfx1250 ASYNCcnt path).
// Each of the 8 waves issues exactly CPY_IT(=16) b128 instructions.
__device__ __forceinline__ void issue_tile_async(const float* __restrict__ Bg,
                                                 float (*Bsb)[WROW], int tid) {
#pragma unroll
  for (int it = 0; it < CPY_IT; ++it) {
    int idx = tid * 4 + it * BLOCK * 4;
    unsigned lds = (unsigned)(uintptr_t)&Bsb[idx >> 7][idx & (WIDTH - 1)];
    const void* g = (const void*)(Bg + idx);
    asm volatile("global_load_async_to_lds_b128 %0, %1, off"
                 :: "v"(lds), "v"(g) : "memory");
  }
}

// --------------------------------------------------------------------------
// Preprocess transition weights:  lw = log_softmax(W, axis=-1)
//   Bmat[t,j,k] = exp(w[j,k] - max_k w[j,k]);  mw[t,j] = -log(sum_k Bmat)
// --------------------------------------------------------------------------
__global__ __launch_bounds__(BLOCK) void hmm_prep_w(const float* __restrict__ W,
                                                    float* __restrict__ Bmat,
                                                    float* __restrict__ mw) {
  int row  = blockIdx.x * (BLOCK / 32) + (threadIdx.x >> 5);
  int lane = threadIdx.x & 31;
  if (row >= STEPS * WIDTH) return;
  const float* wr = W + (size_t)row * WIDTH;
  float4 v = *(const float4*)(wr + lane * 4);
  float m = fmaxf(fmaxf(v.x, v.y), fmaxf(v.z, v.w));
#pragma unroll
  for (int off = 16; off > 0; off >>= 1) m = fmaxf(m, __shfl_xor(m, off, 32));
  float4 e;
  e.x = fast_exp(v.x - m); e.y = fast_exp(v.y - m);
  e.z = fast_exp(v.z - m); e.w = fast_exp(v.w - m);
  float s = e.x + e.y + e.z + e.w;
#pragma unroll
  for (int off = 16; off > 0; off >>= 1) s += __shfl_xor(s, off, 32);
  *(float4*)(Bmat + (size_t)row * WIDTH + lane * 4) = e;
  if (lane == 0) mw[row] = -fast_log(s);
}

// --------------------------------------------------------------------------
// Preprocess leaves: LT[c*128 + j] = log_softmax(input_distros[j,:])[c]
// --------------------------------------------------------------------------
__global__ void hmm_prep_leaf(const float* __restrict__ P, float* __restrict__ LT) {
  int j = threadIdx.x;
  if (j >= WIDTH) return;
  float v[N_BINS];
  float m = -3.402823466e+38f;
#pragma unroll
  for (int c = 0; c < N_BINS; ++c) { v[c] = P[j * N_BINS + c]; m = fmaxf(m, v[c]); }
  float s = 0.f;
#pragma unroll
  for (int c = 0; c < N_BINS; ++c) s += fast_exp(v[c] - m);
  float lse = m + fast_log(s);
#pragma unroll
  for (int c = 0; c < N_BINS; ++c) LT[c * WIDTH + j] = v[c] - lse;
}

// --------------------------------------------------------------------------
// Main forward recurrence: 64 batch rows per workgroup, 255 sequential steps.
// Weight tiles double-buffered in LDS via async global->LDS DMA; per-step
// row-max -> exp (in place) -> fp32 WMMA GEMM -> fused log/add epilogue.
// --------------------------------------------------------------------------
__global__ __launch_bounds__(BLOCK, 1) void hmm_forward(const float* __restrict__ data,
                                                        const float* __restrict__ Bmat,
                                                        const float* __restrict__ mwg,
                                                        const float* __restrict__ LTg,
                                                        float* __restrict__ out) {
  __shared__ __align__(16) float Xs[ROWS][WROW];       // state x / A tile (in place)
  __shared__ __align__(16) float Bs[2][WIDTH][WROW];   // double-buffered weight tile
  __shared__ float LTs[N_BINS * WIDTH];
  __shared__ unsigned char bins_s[ROWS * LENGTH];
  __shared__ float mxs[ROWS];
  __shared__ float mws[WIDTH];

  const int tid  = threadIdx.x;
  const int row0 = blockIdx.x * ROWS;

  const int lane  = tid & 31;
  const int wave  = tid >> 5;
  const int rb    = (wave >> 1) * 16;   // 16-row band
  const int jh    = wave & 1;           // which 64-col half
  const int m16   = lane & 15;
  const int koff  = (lane >> 4) * 2;    // f32 16x16x4 frag: lanes16-31 hold K+2
  const int half8 = (lane >> 4) << 3;   // C/D frag: lanes16-31 hold M+8

  // ---- prologue: async-stage tile 0, leaf table, bins, x0 ----
  issue_tile_async(Bmat, Bs[0], tid);
  for (int idx = tid; idx < N_BINS * WIDTH; idx += BLOCK) LTs[idx] = LTg[idx];
  for (int idx = tid; idx < ROWS * LENGTH; idx += BLOCK) {
    int r = idx >> 8, c = idx & (LENGTH - 1);
    float d = data[(size_t)(row0 + r) * LENGTH + c];
    int b = (int)floorf(d / 0.1f);
    bins_s[idx] = (unsigned char)min(N_BINS - 1, b);
  }
  __syncthreads();
  for (int idx = tid; idx < ROWS * WIDTH; idx += BLOCK) {
    int r = idx >> 7, c = idx & (WIDTH - 1);
    Xs[r][c] = LTs[(int)bins_s[r * LENGTH] * WIDTH + c];
  }
  __syncthreads();

  for (int i = 0; i < STEPS; ++i) {
    const int buf = i & 1;
    // issue next tile's async DMA immediately (overlaps with this whole step)
    if (i + 1 < STEPS)
      issue_tile_async(Bmat + (size_t)(i + 1) * WIDTH * WIDTH, Bs[buf ^ 1], tid);
    if (tid < WIDTH) mws[tid] = mwg[i * WIDTH + tid];
    if (i + 2 < STEPS) {  // warm L2 for the tile after next
      const char* nb = (const char*)(Bmat + (size_t)(i + 2) * WIDTH * WIDTH);
      __builtin_prefetch(nb + tid * 256, 0, 1);
      __builtin_prefetch(nb + tid * 256 + 128, 0, 1);
    }
    // per-row max of current state
    if (tid < ROWS) {
      const float* xr = Xs[tid];
      float m = -3.402823466e+38f;
      for (int c = 0; c < WIDTH; c += 4) {
        float4 v = *(const float4*)(xr + c);
        m = fmaxf(m, fmaxf(fmaxf(v.x, v.y), fmaxf(v.z, v.w)));
      }
      mxs[tid] = m;
    }
    __syncthreads();
    // A[b,k] = exp(x - mx)  (in place)
    for (int idx = tid * 4; idx < ROWS * WIDTH; idx += BLOCK * 4) {
      int r = idx >> 7, c = idx & (WIDTH - 1);
      float mr = mxs[r];
      float4 v = *(const float4*)&Xs[r][c];
      v.x = fast_exp(v.x - mr); v.y = fast_exp(v.y - mr);
      v.z = fast_exp(v.z - mr); v.w = fast_exp(v.w - mr);
      *(float4*)&Xs[r][c] = v;
    }
    // current tile's 16 async ops/wave precede the 16 just issued; async loads
    // retire in order, so <=16 outstanding means Bs[buf] is fully landed.
    if (i + 1 < STEPS) asm volatile("s_wait_asynccnt 16" ::: "memory");
    else               asm volatile("s_wait_asynccnt 0"  ::: "memory");
    __syncthreads();

    // ---- fp32 WMMA GEMM: acc[b,j] = sum_k A[b,k] * Bmat[j,k] ----
    v8f acc[4] = {};
    for (int kk = 0; kk < WIDTH; kk += 4) {
      const float* ap = &Xs[rb + m16][kk + koff];
      v2f a; a.x = ap[0]; a.y = ap[1];
#pragma unroll
      for (int t = 0; t < 4; ++t) {
        const float* bp = &Bs[buf][(jh * 4 + t) * 16 + m16][kk + koff];
        v2f b; b.x = bp[0]; b.y = bp[1];
        acc[t] = __builtin_amdgcn_wmma_f32_16x16x4_f32(
            false, a, false, b, (short)0, acc[t], false, false);
      }
    }
    __syncthreads();   // all A/B LDS reads done before state overwrite

    // ---- epilogue: x_new = mx + mw + log(acc) + emission(step i+1) ----
#pragma unroll
    for (int t = 0; t < 4; ++t) {
      int col = (jh * 4 + t) * 16 + m16;
      float mwv = mws[col];
#pragma unroll
      for (int v = 0; v < 8; ++v) {
        int row = rb + half8 + v;
        float e = LTs[(int)bins_s[row * LENGTH + (i + 1)] * WIDTH + col];
        Xs[row][col] = mxs[row] + mwv + fast_log(acc[t][v]) + e;
      }
    }
    __syncthreads();
  }

  // ---- y[b] = exp(logsumexp_j x[b,j]) ----
  if (tid < ROWS) {
    const float* xr = Xs[tid];
    float m = -3.402823466e+38f;
    for (int c = 0; c < WIDTH; ++c) m = fmaxf(m, xr[c]);
    float s = 0.f;
    for (int c = 0; c < WIDTH; ++c) s += fast_exp(xr[c] - m);
    out[row0 + tid] = s * fast_exp(m);
  }
}

// --------------------------------------------------------------------------
extern "C" void kernel_launch(void* const* d_in, const int* in_sizes, int n_in,
                              void* d_out, int out_size, void* d_ws, size_t ws_size,
                              hipStream_t stream) {
  const float* data    = (const float*)d_in[0];  // (4096, 256)
  const float* distros = (const float*)d_in[1];  // (128, 10)
  const float* Wg      = (const float*)d_in[2];  // (255, 128, 128)

  float* ws   = (float*)d_ws;
  float* Bmat = ws;                                        // 255*128*128
  float* mw   = Bmat + (size_t)STEPS * WIDTH * WIDTH;      // 255*128
  float* LT   = mw + STEPS * WIDTH;                        // 10*128

  hmm_prep_w<<<(STEPS * WIDTH) / (BLOCK / 32), BLOCK, 0, stream>>>(Wg, Bmat, mw);
  hmm_prep_leaf<<<1, WIDTH, 0, stream>>>(distros, LT);
  hmm_forward<<<BATCH / ROWS, BLOCK, 0, stream>>>(data, Bmat, mw, LT, (float*)d_out);
}